// GraphCrossAttention_27178553049903
// MI455X (gfx1250) — compile-verified
//
#include <hip/hip_runtime.h>
#include <hip/hip_bf16.h>

typedef __attribute__((ext_vector_type(2))) float v2f;
typedef __attribute__((ext_vector_type(8))) float v8f;

#define NNODE 110592   // 48^3
#define FEATW 160      // [xl_q(8) xr_q(8) xl_k(8) xr_k(8) xl_v(64) xr_v(64)]
#define NEGINF (-1e30f)

__device__ __forceinline__ v8f wmma_f32(v2f a, v2f b, v8f c) {
  // D = A(16x4 f32) * B(4x16 f32) + C(16x16 f32)
  return __builtin_amdgcn_wmma_f32_16x16x4_f32(false, a, false, b, (short)0, c,
                                               false, false);
}

// ---------------------------------------------------------------------------
// Stage 0a: pack the six GATv2 weight matrices into Wcat[64][160]
// ---------------------------------------------------------------------------
__global__ void k_pack_w(const float* __restrict__ qwl, const float* __restrict__ qwr,
                         const float* __restrict__ kwl, const float* __restrict__ kwr,
                         const float* __restrict__ vwl, const float* __restrict__ vwr,
                         float* __restrict__ W) {
  int id = blockIdx.x * blockDim.x + threadIdx.x;   // 64*160 exact
  int k = id / FEATW, c = id % FEATW;
  float v;
  if (c < 8)       v = qwl[k * 8 + c];
  else if (c < 16) v = qwr[k * 8 + c - 8];
  else if (c < 24) v = kwl[k * 8 + c - 16];
  else if (c < 32) v = kwr[k * 8 + c - 24];
  else if (c < 96) v = vwl[k * 64 + c - 32];
  else             v = vwr[k * 64 + c - 96];
  W[id] = v;
}

// ---------------------------------------------------------------------------
// Stage 0b: Weff[64][128] = [gamma*W1 ; W1+W2]  so that
// merge becomes  out = Weff @ [osum ; x] + b   (== W @ [gamma*osum+x ; x] + b)
// ---------------------------------------------------------------------------
__global__ void k_pack_merge(const float* __restrict__ mw, const float* __restrict__ gamma,
                             float* __restrict__ weff) {
  int id = blockIdx.x * blockDim.x + threadIdx.x;   // 64*128 exact
  int o = id / 128, k = id % 128;
  float v;
  if (k < 64) v = gamma[0] * mw[o * 128 + k];
  else        v = mw[o * 128 + (k - 64)] + mw[o * 128 + k];
  weff[id] = v;
}

// ---------------------------------------------------------------------------
// Stage 1: feat[N,160] = X[N,64] @ Wcat[64,160].
// One wave per 16-row M tile, 10 accumulators covering all of N=160; A (X)
// read exactly once (global_load_b64). Wcat staged in LDS *pair-interleaved*
// (sWp[k/2][n][2] = {W[k][n], W[k+1][n]}) so every B operand is one aligned
// ds_load_b64 directly into the WMMA register pair (no repack movs).
// ---------------------------------------------------------------------------
__global__ void k_gemm_feat(const float* __restrict__ X, const float* __restrict__ W,
                            float* __restrict__ F) {
  __shared__ float sWp[64 * FEATW];                         // 40 KB
  for (int i = threadIdx.x; i < 64 * FEATW; i += 128) {
    int k = i / FEATW, n = i % FEATW;
    sWp[(k >> 1) * (2 * FEATW) + n * 2 + (k & 1)] = W[i];
  }
  __syncthreads();
  int wid  = blockIdx.x * 4 + (threadIdx.x >> 5);           // 6912 M tiles exact
  int lane = threadIdx.x & 31, l16 = lane & 15, hh = lane >> 4;
  const float* arow = X + (size_t)(wid * 16 + l16) * 64;
  v8f acc[10];
#pragma unroll
  for (int t = 0; t < 10; ++t) acc[t] = (v8f){0.f,0.f,0.f,0.f,0.f,0.f,0.f,0.f};
#pragma unroll
  for (int k = 0; k < 64; k += 4) {
    int ka = k + 2 * hh;                                    // always even
    v2f a = *(const v2f*)&arow[ka];
    const float* bp = &sWp[(ka >> 1) * (2 * FEATW)];
#pragma unroll
    for (int tn = 0; tn < 10; ++tn) {
      v2f b = *(const v2f*)&bp[(tn * 16 + l16) * 2];
      acc[tn] = wmma_f32(a, b, acc[tn]);
    }
  }
#pragma unroll
  for (int tn = 0; tn < 10; ++tn)
#pragma unroll
    for (int r = 0; r < 8; ++r)
      F[(size_t)(wid * 16 + hh * 8 + r) * FEATW + tn * 16 + l16] = acc[tn][r];
}

// ---------------------------------------------------------------------------
// Stage 2: GATv2 (q,k,v heads) per node; neighbors derived from the 6-connected
// grid + self-loop (exactly the reference edge set), so no atomics needed.
// ---------------------------------------------------------------------------
__global__ void k_gat(const float* __restrict__ feat,
                      const float* __restrict__ qatt, const float* __restrict__ qb,
                      const float* __restrict__ katt, const float* __restrict__ kb,
                      const float* __restrict__ vatt, const float* __restrict__ vb,
                      float* __restrict__ pq, float* __restrict__ pk,
                      float* __restrict__ pv) {
  int i = blockIdx.x * blockDim.x + threadIdx.x;    // 110592 exact
  int w = i % 48, h = (i / 48) % 48, d = i / 2304;
  int nbr[7]; int nn = 0;
  nbr[nn++] = i;
  if (w > 0)  nbr[nn++] = i - 1;
  if (w < 47) nbr[nn++] = i + 1;
  if (h > 0)  nbr[nn++] = i - 48;
  if (h < 47) nbr[nn++] = i + 48;
  if (d > 0)  nbr[nn++] = i - 2304;
  if (d < 47) nbr[nn++] = i + 2304;
  const float* fi = feat + (size_t)i * FEATW;
  float lq[7], lk[7], lv[7];
  for (int e = 0; e < nn; ++e) {
    const float* fj = feat + (size_t)nbr[e] * FEATW;
    float sq = 0.f, sk = 0.f, sv = 0.f;
    for (int c = 0; c < 8; ++c) {
      float t = fj[c]      + fi[8 + c];  t = t > 0.f ? t : 0.2f * t; sq += qatt[c] * t;
      float u = fj[16 + c] + fi[24 + c]; u = u > 0.f ? u : 0.2f * u; sk += katt[c] * u;
    }
    for (int c = 0; c < 64; ++c) {
      float t = fj[32 + c] + fi[96 + c]; t = t > 0.f ? t : 0.2f * t; sv += vatt[c] * t;
    }
    lq[e] = sq; lk[e] = sk; lv[e] = sv;
  }
  float mq = lq[0], mk = lk[0], mv = lv[0];
  for (int e = 1; e < nn; ++e) {
    mq = fmaxf(mq, lq[e]); mk = fmaxf(mk, lk[e]); mv = fmaxf(mv, lv[e]);
  }
  float zq = 0.f, zk = 0.f, zv = 0.f;
  for (int e = 0; e < nn; ++e) {
    lq[e] = expf(lq[e] - mq); zq += lq[e];
    lk[e] = expf(lk[e] - mk); zk += lk[e];
    lv[e] = expf(lv[e] - mv); zv += lv[e];
  }
  float iq = 1.f / zq, ik = 1.f / zk, iv = 1.f / zv;
  for (int c = 0; c < 8; ++c) {
    float sq = 0.f, sk = 0.f;
    for (int e = 0; e < nn; ++e) {
      const float* fj = feat + (size_t)nbr[e] * FEATW;
      sq += lq[e] * fj[c];
      sk += lk[e] * fj[16 + c];
    }
    pq[(size_t)i * 8 + c] = sq * iq + qb[c];
    pk[(size_t)i * 8 + c] = sk * ik + kb[c];
  }
  for (int c = 0; c < 64; ++c) {
    float sv = 0.f;
    for (int e = 0; e < nn; ++e)
      sv += lv[e] * feat[(size_t)nbr[e] * FEATW + 32 + c];
    pv[(size_t)i * 64 + c] = sv * iv + vb[c];
  }
}

// ---------------------------------------------------------------------------
// Stage 3: rearrange pq/pk into per-axis GEMM operands.
// qmat[axis][alpha][gamma][beta] (48x48x384, row-major over beta).
// kmat pair-interleaved: kmat[axis][alpha][beta/2][gamma][2] so that egemm's
// B operand {K[beta][g], K[beta+1][g]} is one aligned global_load_b64.
// Index map: g = alpha*384+beta; c=g/2304; r=g%2304;
//   H: h=r/48, w=r%48, d=gamma ; W: d=r/48, w=r%48, h=gamma ; D: w=r/48, h=r%48, d=gamma
// ---------------------------------------------------------------------------
__global__ void k_rearrange_qk(const float* __restrict__ pq, const float* __restrict__ pk,
                               float* __restrict__ qmat, float* __restrict__ kmat) {
  int id = blockIdx.x * blockDim.x + threadIdx.x;   // 884736 exact
  int axis = blockIdx.y;
  int beta = id % 384, gamma = (id / 384) % 48, alpha = id / (384 * 48);
  int g = alpha * 384 + beta;
  int c = g / 2304, r = g % 2304;
  int d, h, w;
  if (axis == 0)      { h = r / 48; w = r % 48; d = gamma; }
  else if (axis == 1) { d = r / 48; w = r % 48; h = gamma; }
  else                { w = r / 48; h = r % 48; d = gamma; }
  int src = ((d * 48 + h) * 48 + w) * 8 + c;
  size_t ao = (size_t)axis * 884736;
  qmat[ao + (size_t)(alpha * 48 + gamma) * 384 + beta] = pq[src];
  kmat[ao + (size_t)(alpha * 192 + (beta >> 1)) * 96 + gamma * 2 + (beta & 1)] = pk[src];
}

// ---------------------------------------------------------------------------
// Stage 4: E[alpha] = Q[alpha](48x384) @ K[alpha](384x48), written straight
// into the concatenated logit buffer att[48][48][144] (H gets the -inf diag).
// One wave per (alpha, m-tile) with 3 accumulators across N=48.
// ---------------------------------------------------------------------------
__global__ void k_egemm(const float* __restrict__ qmat, const float* __restrict__ kmat,
                        float* __restrict__ att) {
  int wid  = blockIdx.x * 4 + (threadIdx.x >> 5);   // 144 waves per axis exact
  int axis = blockIdx.y;
  int lane = threadIdx.x & 31, l16 = lane & 15, hh = lane >> 4;
  int alpha = wid / 3, tm = wid % 3;
  size_t ao = (size_t)axis * 884736;
  const float* A = qmat + ao + (size_t)(alpha * 48 + tm * 16 + l16) * 384;
  const float* B = kmat + ao + (size_t)alpha * 192 * 96;   // pair-interleaved
  v8f acc[3];
#pragma unroll
  for (int t = 0; t < 3; ++t) acc[t] = (v8f){0.f,0.f,0.f,0.f,0.f,0.f,0.f,0.f};
  for (int k = 0; k < 384; k += 4) {
    int ka = k + 2 * hh;                                   // always even
    v2f a = *(const v2f*)&A[ka];
    const float* bp = B + (size_t)(ka >> 1) * 96;
#pragma unroll
    for (int tn = 0; tn < 3; ++tn) {
      v2f b = *(const v2f*)&bp[(tn * 16 + l16) * 2];
      acc[tn] = wmma_f32(a, b, acc[tn]);
    }
  }
#pragma unroll
  for (int tn = 0; tn < 3; ++tn) {
    int n = tn * 16 + l16;
#pragma unroll
    for (int r = 0; r < 8; ++r) {
      int mrow = tm * 16 + hh * 8 + r;
      float v = acc[tn][r];
      if (axis == 0) {
        if (mrow == n) v += NEGINF;                      // diagonal mask on e_H
        att[(size_t)(mrow * 48 + alpha) * 144 + n] = v;  // att[g1][a][k]
      } else if (axis == 1) {
        att[(size_t)(alpha * 48 + mrow) * 144 + 48 + n] = v;
      } else {
        att[(size_t)(alpha * 48 + mrow) * 144 + 96 + n] = v;
      }
    }
  }
}

// Stage 5: softmax over the 144-wide concatenated logit rows (2304 rows).
__global__ void k_softmax(float* __restrict__ att) {
  int row = blockIdx.x * blockDim.x + threadIdx.x;   // 2304 exact
  float* p = att + (size_t)row * 144;
  float m = p[0];
  for (int i = 1; i < 144; ++i) m = fmaxf(m, p[i]);
  float z = 0.f;
  for (int i = 0; i < 144; ++i) { float e = expf(p[i] - m); p[i] = e; z += e; }
  float inv = 1.f / z;
  for (int i = 0; i < 144; ++i) p[i] *= inv;
}

// ---------------------------------------------------------------------------
// Stage 6: O[alpha][m][p] = sum_k pv_axis[alpha][m][k] * a_axis[alpha][p][k]
// (M=3072, N=48, K=48). pv_axis is gathered from pv on the fly (per-lane base
// + constant stride along k), read exactly once per axis thanks to the three
// N-tile accumulators; B pairs are contiguous in att -> single b64 loads.
// Results accumulate into channel-major osum.
// ---------------------------------------------------------------------------
__global__ void k_ogemm(const float* __restrict__ pv, const float* __restrict__ att,
                        float* __restrict__ osum, int axis, int accum) {
  int wid  = blockIdx.x * 4 + (threadIdx.x >> 5);   // 9216 waves exact
  int lane = threadIdx.x & 31, l16 = lane & 15, hh = lane >> 4;
  int alpha = wid / 192, tm = wid % 192;
  int m = tm * 16 + l16;
  int g = alpha * 3072 + m;
  int cg = g / 2304, rr = g % 2304;
  size_t abase; int astr;
  if (axis == 0)      { abase = (size_t)(rr / 48) * 3072   + (rr % 48) * 64 + cg; astr = 147456; }
  else if (axis == 1) { abase = (size_t)(rr / 48) * 147456 + (rr % 48) * 64 + cg; astr = 3072;   }
  else                { abase = (size_t)(rr % 48) * 3072   + (rr / 48) * 64 + cg; astr = 147456; }
  const float* Bp[3];
#pragma unroll
  for (int tn = 0; tn < 3; ++tn) {
    int n = tn * 16 + l16;
    if (axis == 0) Bp[tn] = att + (size_t)(n * 48 + alpha) * 144;      // a_H[alpha][n][k]
    else           Bp[tn] = att + (size_t)(alpha * 48 + n) * 144 + (axis == 1 ? 48 : 96);
  }
  v8f acc[3];
#pragma unroll
  for (int t = 0; t < 3; ++t) acc[t] = (v8f){0.f,0.f,0.f,0.f,0.f,0.f,0.f,0.f};
#pragma unroll
  for (int k = 0; k < 48; k += 4) {
    int ka = k + 2 * hh;                                   // always even
    v2f a;
    a.x = pv[abase + (size_t)ka * astr];
    a.y = pv[abase + (size_t)(ka + 1) * astr];
#pragma unroll
    for (int tn = 0; tn < 3; ++tn) {
      v2f b = *(const v2f*)&Bp[tn][ka];                    // contiguous pair
      acc[tn] = wmma_f32(a, b, acc[tn]);
    }
  }
#pragma unroll
  for (int tn = 0; tn < 3; ++tn) {
    int n = tn * 16 + l16;
#pragma unroll
    for (int r = 0; r < 8; ++r) {
      int mrow = tm * 16 + hh * 8 + r;
      size_t idx;
      if (axis == 0) idx = (size_t)(mrow / 48) * 110592 + (mrow % 48) * 2304 + alpha * 48 + n;
      else           idx = (size_t)(mrow / 48) * 110592 + (mrow % 48) * 2304 + n * 48 + alpha;
      if (accum) osum[idx] += acc[tn][r];
      else       osum[idx]  = acc[tn][r];
    }
  }
}

// ---------------------------------------------------------------------------
// Stage 7: out[64,110592] = Weff[64,128] @ [osum ; x] + merge_b.
// One wave per N tile covers all of M=64 with 4 accumulators, so B (osum,x)
// is streamed exactly once. Weff staged in LDS with pitch 130; A pairs are
// contiguous & 8B-aligned (ka even) -> single ds_load_b64 each.
// ---------------------------------------------------------------------------
__global__ void k_merge(const float* __restrict__ X, const float* __restrict__ osum,
                        const float* __restrict__ weff, const float* __restrict__ mb,
                        float* __restrict__ out) {
  __shared__ float sWe[64 * 130];                           // 33.3 KB
  for (int i = threadIdx.x; i < 64 * 128; i += 128)
    sWe[(i / 128) * 130 + (i % 128)] = weff[i];
  __syncthreads();
  int wid  = blockIdx.x * 4 + (threadIdx.x >> 5);           // 6912 N tiles exact
  int lane = threadIdx.x & 31, l16 = lane & 15, hh = lane >> 4;
  int n = wid * 16 + l16;
  v8f acc[4];
#pragma unroll
  for (int t = 0; t < 4; ++t) acc[t] = (v8f){0.f,0.f,0.f,0.f,0.f,0.f,0.f,0.f};
#pragma unroll
  for (int k = 0; k < 128; k += 4) {
    int ka = k + 2 * hh;               // ka even -> (ka,ka+1) never straddle 64
    v2f b;
    if (ka < 64) {
      b.x = osum[(size_t)ka * 110592 + n];
      b.y = osum[(size_t)(ka + 1) * 110592 + n];
    } else {
      b.x = X[(size_t)(ka - 64) * 110592 + n];
      b.y = X[(size_t)(ka - 63) * 110592 + n];
    }
#pragma unroll
    for (int tm = 0; tm < 4; ++tm) {
      v2f a = *(const v2f*)&sWe[(tm * 16 + l16) * 130 + ka];   // even index -> 8B aligned
      acc[tm] = wmma_f32(a, b, acc[tm]);
    }
  }
#pragma unroll
  for (int tm = 0; tm < 4; ++tm)
#pragma unroll
    for (int r = 0; r < 8; ++r) {
      int mrow = tm * 16 + hh * 8 + r;
      out[(size_t)mrow * 110592 + n] = acc[tm][r] + mb[mrow];
    }
}

// ---------------------------------------------------------------------------
extern "C" void kernel_launch(void* const* d_in, const int* in_sizes, int n_in,
                              void* d_out, int out_size, void* d_ws, size_t ws_size,
                              hipStream_t stream) {
  (void)in_sizes; (void)n_in; (void)out_size; (void)ws_size;
  const float* x    = (const float*)d_in[0];
  const float* qwl  = (const float*)d_in[1];
  const float* qwr  = (const float*)d_in[2];
  const float* qatt = (const float*)d_in[3];
  const float* qb   = (const float*)d_in[4];
  const float* kwl  = (const float*)d_in[5];
  const float* kwr  = (const float*)d_in[6];
  const float* katt = (const float*)d_in[7];
  const float* kb   = (const float*)d_in[8];
  const float* vwl  = (const float*)d_in[9];
  const float* vwr  = (const float*)d_in[10];
  const float* vatt = (const float*)d_in[11];
  const float* vb   = (const float*)d_in[12];
  const float* gmma = (const float*)d_in[13];
  const float* mw   = (const float*)d_in[14];
  const float* mb   = (const float*)d_in[15];
  float* out = (float*)d_out;

  float* ws   = (float*)d_ws;
  float* wcat = ws;                  // 10,240
  float* weff = wcat + 10240;        // 8,192
  float* feat = weff + 8192;         // 17,694,720
  float* pq   = feat + 17694720;     // 884,736
  float* pk   = pq   + 884736;       // 884,736
  float* pv   = pk   + 884736;       // 7,077,888
  float* qmat = pv   + 7077888;      // 2,654,208
  float* kmat = qmat + 2654208;      // 2,654,208  (pair-interleaved)
  float* att  = kmat + 2654208;      // 331,776
  float* osum = att  + 331776;       // 7,077,888   (total ~157 MB)

  k_pack_w<<<40, 256, 0, stream>>>(qwl, qwr, kwl, kwr, vwl, vwr, wcat);
  k_pack_merge<<<32, 256, 0, stream>>>(mw, gmma, weff);
  k_gemm_feat<<<1728, 128, 0, stream>>>(x, wcat, feat);           // 6912 waves
  k_gat<<<432, 256, 0, stream>>>(feat, qatt, qb, katt, kb, vatt, vb, pq, pk, pv);
  k_rearrange_qk<<<dim3(3456, 3), 256, 0, stream>>>(pq, pk, qmat, kmat);
  k_egemm<<<dim3(36, 3), 128, 0, stream>>>(qmat, kmat, att);      // 144 waves/axis
  k_softmax<<<9, 256, 0, stream>>>(att);
  k_ogemm<<<2304, 128, 0, stream>>>(pv, att, osum, 0, 0);         // store
  k_ogemm<<<2304, 128, 0, stream>>>(pv, att, osum, 1, 1);         // accumulate
  k_ogemm<<<2304, 128, 0, stream>>>(pv, att, osum, 2, 1);         // accumulate
  k_merge<<<1728, 128, 0, stream>>>(x, osum, weff, mb, out);
}